// Transformer_42932493091037
// MI455X (gfx1250) — compile-verified
//
#include <hip/hip_runtime.h>
#include <math.h>

typedef __attribute__((ext_vector_type(2))) float v2f;
typedef __attribute__((ext_vector_type(4))) float v4f;
typedef __attribute__((ext_vector_type(8))) float v8f;

#define EMBED 16
#define SEQ 8
#define FFD 32
#define VOCAB 256
#define WAVES 8
#define SCR_STRIDE 17   // pad 16x16 tile rows to avoid LDS bank conflicts

// ---- WMMA helpers: D = A(16x16) * B(16x16) + C via 4 chained 16x16x4 f32 WMMAs ----
__device__ __forceinline__ v8f wmma16(const v2f a[4], const v2f b[4], v8f c) {
#pragma unroll
  for (int s = 0; s < 4; ++s)
    c = __builtin_amdgcn_wmma_f32_16x16x4_f32(
        /*neg_a=*/false, a[s], /*neg_b=*/false, b[s],
        /*c_mod=*/(short)0, c, /*reuse_a=*/false, /*reuse_b=*/false);
  return c;
}

// C/D layout: VGPR r -> (M = rbase + r, N = col); store tile to LDS scratch.
__device__ __forceinline__ void store_C(float* scr, const v8f& c, int col, int rbase) {
#pragma unroll
  for (int r = 0; r < 8; ++r) scr[(rbase + r) * SCR_STRIDE + col] = c[r];
}

// A layout (16x4 slice s): lane m (0-15): K = 4s+{0,1}; lane m+16: K = 4s+{2,3}
__device__ __forceinline__ void load_A(const float* scr, v2f a[4], int lane) {
  int m = lane & 15;
  int koff = (lane >> 4) << 1;
#pragma unroll
  for (int s = 0; s < 4; ++s) {
    a[s].x = scr[m * SCR_STRIDE + 4 * s + koff];
    a[s].y = scr[m * SCR_STRIDE + 4 * s + koff + 1];
  }
}

// B layout (4x16 slice s) from row-major [K x N] with row stride ldb:
// lane n (0-15): rows K = 4s+{0,1}; lane n+16: rows K = 4s+{2,3}
__device__ __forceinline__ void load_B(const float* W, int ldb, v2f b[4], int lane) {
  int n = lane & 15;
  int kk = (lane >> 4) << 1;
#pragma unroll
  for (int s = 0; s < 4; ++s) {
    b[s].x = W[(4 * s + kk) * ldb + n];
    b[s].y = W[(4 * s + kk + 1) * ldb + n];
  }
}

__global__ __launch_bounds__(256) void tformer_fwd(
    const int* __restrict__ x,
    const float* __restrict__ tok, const float* __restrict__ pos,
    const float* __restrict__ Wq, const float* __restrict__ bq,
    const float* __restrict__ Wk, const float* __restrict__ bk,
    const float* __restrict__ Wv, const float* __restrict__ bv,
    const float* __restrict__ Wo, const float* __restrict__ bo,
    const float* __restrict__ W1, const float* __restrict__ b1,
    const float* __restrict__ W2, const float* __restrict__ b2,
    const float* __restrict__ Wh, const float* __restrict__ bh,
    float* __restrict__ out, int Btot)
{
  __shared__ float s_tok[VOCAB * EMBED];            // 16 KB
  __shared__ float s_pos[SEQ * EMBED];              // 0.5 KB
  __shared__ float s_wh[EMBED * VOCAB];             // 16 KB, row-major [k][n]
  __shared__ float s_scr[WAVES][16 * SCR_STRIDE];   // layout-conversion scratch
  __shared__ float s_stage[WAVES][16 * 16];         // output staging (16B-aligned rows)

  const int tid = threadIdx.x;
  // vectorized one-time preloads (tables are 16B-aligned, counts divisible by 4)
  {
    const v4f* tok4 = (const v4f*)tok;
    const v4f* wh4  = (const v4f*)Wh;
    const v4f* pos4 = (const v4f*)pos;
    v4f* stok4 = (v4f*)s_tok;
    v4f* swh4  = (v4f*)s_wh;
    v4f* spos4 = (v4f*)s_pos;
    for (int i = tid; i < (VOCAB * EMBED) / 4; i += 256) stok4[i] = tok4[i];
    for (int i = tid; i < (EMBED * VOCAB) / 4; i += 256) swh4[i] = wh4[i];
    for (int i = tid; i < (SEQ * EMBED) / 4; i += 256) spos4[i] = pos4[i];
  }
  __syncthreads();

  const int lane  = tid & 31;
  const int wave  = tid >> 5;
  const int pair  = blockIdx.x * WAVES + wave;   // 2 sequences per wave
  if (pair * 2 + 1 >= Btot) return;
  const int col   = lane & 15;
  const int rbase = (lane >> 4) * 8;
  float* scr   = &s_scr[wave][0];
  float* stage = &s_stage[wave][0];

  // ---- X = token_embed[x] + pos_embed  (16 rows = 2 seqs x 8 tokens) ----
  v8f xc;
  {
    const int seq = pair * 2 + (rbase >> 3);
    const int* xrow = x + seq * SEQ;
#pragma unroll
    for (int r = 0; r < 8; ++r) {
      int id = xrow[r];
      float v = s_tok[id * EMBED + col] + s_pos[r * EMBED + col];
      xc[r] = v;
      scr[(rbase + r) * SCR_STRIDE + col] = v;
    }
  }
  v2f xa[4];
  load_A(scr, xa, lane);

  // ---- Q, K, V projections ----
  v2f wb[4];
  v8f q, k, v;
  {
    float bql = bq[col], bkl = bk[col], bvl = bv[col];
#pragma unroll
    for (int r = 0; r < 8; ++r) { q[r] = bql; k[r] = bkl; v[r] = bvl; }
  }
  load_B(Wq, EMBED, wb, lane); q = wmma16(xa, wb, q);
  load_B(Wk, EMBED, wb, lane); k = wmma16(xa, wb, k);
  load_B(Wv, EMBED, wb, lane); v = wmma16(xa, wb, v);

  // ---- layout conversions through LDS (same-wave DS ops are in-order) ----
  v2f qa[4], ka[4], vb[4];
  store_C(scr, q, col, rbase); load_A(scr, qa, lane);
  store_C(scr, k, col, rbase); load_A(scr, ka, lane);  // A-layout(K) == B-layout(K^T)
  store_C(scr, v, col, rbase); load_B(scr, SCR_STRIDE, vb, lane);

  // ---- scores = Q K^T / sqrt(D), block-diagonal masked softmax ----
  v8f sc;
#pragma unroll
  for (int r = 0; r < 8; ++r) sc[r] = 0.f;
  sc = wmma16(qa, ka, sc);

  const bool valid = (col < 8) == (rbase == 0);  // same-sequence 8x8 blocks only
  v8f w;
#pragma unroll
  for (int r = 0; r < 8; ++r) {
    float sv = valid ? sc[r] * 0.25f : -1e30f;
    float mx = sv;
    mx = fmaxf(mx, __shfl_xor(mx, 1));
    mx = fmaxf(mx, __shfl_xor(mx, 2));
    mx = fmaxf(mx, __shfl_xor(mx, 4));
    float p = valid ? __expf(sv - mx) : 0.f;
    float sum = p;
    sum += __shfl_xor(sum, 1);
    sum += __shfl_xor(sum, 2);
    sum += __shfl_xor(sum, 4);
    w[r] = valid ? p / sum : 0.f;
  }

  // ---- attn = weights @ V ----
  v2f wa[4];
  store_C(scr, w, col, rbase); load_A(scr, wa, lane);
  v8f at;
#pragma unroll
  for (int r = 0; r < 8; ++r) at[r] = 0.f;
  at = wmma16(wa, vb, at);

  // ---- X1 = X + attn @ Wo + bo ----
  v2f oa[4];
  store_C(scr, at, col, rbase); load_A(scr, oa, lane);
  load_B(Wo, EMBED, wb, lane);
  v8f x1;
  {
    float bol = bo[col];
#pragma unroll
    for (int r = 0; r < 8; ++r) x1[r] = xc[r] + bol;
  }
  x1 = wmma16(oa, wb, x1);

  // ---- FF: H = relu(X1 @ W1 + b1)  (W1 is 16x32 -> two N tiles) ----
  v2f x1a[4];
  store_C(scr, x1, col, rbase); load_A(scr, x1a, lane);
  v8f h0, h1;
  {
    float b1l0 = b1[col], b1l1 = b1[16 + col];
#pragma unroll
    for (int r = 0; r < 8; ++r) { h0[r] = b1l0; h1[r] = b1l1; }
  }
  load_B(W1, FFD, wb, lane);      h0 = wmma16(x1a, wb, h0);
  load_B(W1 + 16, FFD, wb, lane); h1 = wmma16(x1a, wb, h1);
#pragma unroll
  for (int r = 0; r < 8; ++r) { h0[r] = fmaxf(h0[r], 0.f); h1[r] = fmaxf(h1[r], 0.f); }

  // ---- X2 = X1 + H @ W2 + b2   (W2 is 32x16 -> two K tiles) ----
  v8f x2;
  {
    float b2l = b2[col];
#pragma unroll
    for (int r = 0; r < 8; ++r) x2[r] = x1[r] + b2l;
  }
  v2f ha[4];
  store_C(scr, h0, col, rbase); load_A(scr, ha, lane);
  load_B(W2, EMBED, wb, lane);              x2 = wmma16(ha, wb, x2);
  store_C(scr, h1, col, rbase); load_A(scr, ha, lane);
  load_B(W2 + 16 * EMBED, EMBED, wb, lane); x2 = wmma16(ha, wb, x2);

  // ---- logits = X2 @ W_head + b_head  (16 N tiles of 16) ----
  v2f x2a[4];
  store_C(scr, x2, col, rbase); load_A(scr, x2a, lane);
  const size_t obase = (size_t)pair * 16 * VOCAB;
  // lane -> two float4 quads of the staged 16x16 tile: row = lane/2, cols (lane&1)*8 + {0,4}
  const int srow = lane >> 1;
  const int scol = (lane & 1) * 8;
  for (int nt = 0; nt < 16; ++nt) {
    v2f bt[4];
    load_B(s_wh + nt * 16, VOCAB, bt, lane);
    float bhl = bh[nt * 16 + col];
    v8f acc;
#pragma unroll
    for (int r = 0; r < 8; ++r) acc[r] = bhl;
    acc = wmma16(x2a, bt, acc);
    // stage C-layout tile in LDS, then emit two coalesced b128 stores per lane
#pragma unroll
    for (int r = 0; r < 8; ++r) stage[(rbase + r) * 16 + col] = acc[r];
    v4f q0 = *(const v4f*)&stage[srow * 16 + scol];
    v4f q1 = *(const v4f*)&stage[srow * 16 + scol + 4];
    float* orow = out + obase + (size_t)srow * VOCAB + nt * 16 + scol;
    *(v4f*)orow = q0;
    *(v4f*)(orow + 4) = q1;
  }
}

extern "C" void kernel_launch(void* const* d_in, const int* in_sizes, int n_in,
                              void* d_out, int out_size, void* d_ws, size_t ws_size,
                              hipStream_t stream) {
  (void)n_in; (void)out_size; (void)d_ws; (void)ws_size;
  const int*   x    = (const int*)d_in[0];
  const float* tok  = (const float*)d_in[1];
  const float* pos  = (const float*)d_in[2];
  const float* Wq   = (const float*)d_in[3];
  const float* bq   = (const float*)d_in[4];
  const float* Wk   = (const float*)d_in[5];
  const float* bk   = (const float*)d_in[6];
  const float* Wv   = (const float*)d_in[7];
  const float* bv   = (const float*)d_in[8];
  const float* Wo   = (const float*)d_in[9];
  const float* bo   = (const float*)d_in[10];
  const float* W1   = (const float*)d_in[11];
  const float* b1   = (const float*)d_in[12];
  const float* W2   = (const float*)d_in[13];
  const float* b2   = (const float*)d_in[14];
  const float* Wh   = (const float*)d_in[15];
  const float* bh   = (const float*)d_in[16];
  float* out = (float*)d_out;

  const int Btot = in_sizes[0] / SEQ;                 // 32768
  const int blocks = (Btot / 2 + WAVES - 1) / WAVES;  // 2 seqs/wave, 8 waves/block
  tformer_fwd<<<blocks, 256, 0, stream>>>(x, tok, pos, Wq, bq, Wk, bk, Wv, bv,
                                          Wo, bo, W1, b1, W2, b2, Wh, bh, out, Btot);
}